// Jitter_24352464568637
// MI455X (gfx1250) — compile-verified
//
#include <hip/hip_runtime.h>
#include <stdint.h>

// Jitter: out[b,c,t] = in[b,c,idx[t]], idx[t] in {t-1, t, t+1} chosen per
// timestep by JAX threefry-derived Bernoulli draws (seed 42).
// Pure bandwidth problem: 0 FLOPs, ~0.54 GB traffic -> ~23 us floor @ 23.3 TB/s.

#define JITTER_T 8192          // time dimension
#define HALF_T   (JITTER_T / 2)

// ---------------------------------------------------------------------------
// Threefry-2x32 (JAX-compatible: 20 rounds, inject every 4)
// ---------------------------------------------------------------------------
__device__ __forceinline__ uint32_t rotl32(uint32_t x, int n) {
  return (x << n) | (x >> (32 - n));
}

__device__ __forceinline__ void threefry2x32(uint32_t k0, uint32_t k1,
                                             uint32_t x0, uint32_t x1,
                                             uint32_t &o0, uint32_t &o1) {
  const uint32_t ks0 = k0;
  const uint32_t ks1 = k1;
  const uint32_t ks2 = k0 ^ k1 ^ 0x1BD11BDAu;
  x0 += ks0; x1 += ks1;
#define TF_ROUND(r) { x0 += x1; x1 = rotl32(x1, (r)); x1 ^= x0; }
  TF_ROUND(13) TF_ROUND(15) TF_ROUND(26) TF_ROUND(6)
  x0 += ks1; x1 += ks2 + 1u;
  TF_ROUND(17) TF_ROUND(29) TF_ROUND(16) TF_ROUND(24)
  x0 += ks2; x1 += ks0 + 2u;
  TF_ROUND(13) TF_ROUND(15) TF_ROUND(26) TF_ROUND(6)
  x0 += ks0; x1 += ks1 + 3u;
  TF_ROUND(17) TF_ROUND(29) TF_ROUND(16) TF_ROUND(24)
  x0 += ks1; x1 += ks2 + 4u;
  TF_ROUND(13) TF_ROUND(15) TF_ROUND(26) TF_ROUND(6)
  x0 += ks2; x1 += ks0 + 5u;
#undef TF_ROUND
  o0 = x0; o1 = x1;
}

__device__ __forceinline__ float u01(uint32_t w) {
  union { uint32_t u; float f; } c;
  c.u = 0x3F800000u | (w >> 9);   // [1,2)
  return c.f - 1.0f;              // [0,1), matches jax._src.random._uniform
}

// ---------------------------------------------------------------------------
// Kernel 1: build idx[8192] into workspace (runs once per launch; tiny).
// Pair p handles timesteps t0=p (output word0) and t1=p+4096 (output word1),
// matching JAX's split-counter layout for shape (8192,).
// ---------------------------------------------------------------------------
__global__ void jitter_idx_kernel(int* __restrict__ idx) {
  const int p = blockIdx.x * blockDim.x + threadIdx.x;
  if (p >= HALF_T) return;

  // k_rep, k_dir = jax.random.split(jax.random.key(42)):
  // threefry(key=(0,42), counts=iota(4)) with x0=[0,1], x1=[2,3];
  // k_rep = (y0(pair(0,2)), y0(pair(1,3))), k_dir = (y1(pair(0,2)), y1(pair(1,3)))
  uint32_t a0, b0, a1, b1;
  threefry2x32(0u, 42u, 0u, 2u, a0, b0);
  threefry2x32(0u, 42u, 1u, 3u, a1, b1);
  const uint32_t kr0 = a0, kr1 = a1;   // replace key
  const uint32_t kd0 = b0, kd1 = b1;   // direction key

  uint32_t r0, r1, d0, d1;
  threefry2x32(kr0, kr1, (uint32_t)p, (uint32_t)(p + HALF_T), r0, r1);
  threefry2x32(kd0, kd1, (uint32_t)p, (uint32_t)(p + HALF_T), d0, d1);

  // t0 = p (< 4096, never T-1)
  {
    const int t = p;
    const bool rep = u01(r0) < 0.12f;
    int dir = (u01(d0) < 0.5f) ? 1 : -1;
    if (t == 0) dir = 1;
    idx[t] = rep ? (t + dir) : t;
  }
  // t1 = p + 4096 (never 0)
  {
    const int t = p + HALF_T;
    const bool rep = u01(r1) < 0.12f;
    int dir = (u01(d1) < 0.5f) ? 1 : -1;
    if (t == JITTER_T - 1) dir = -1;
    idx[t] = rep ? (t + dir) : t;
  }
}

// ---------------------------------------------------------------------------
// Kernel 2: one block per (b,c) row. Stage the 32KB row into LDS with CDNA5
// async B128 global->LDS transfers (ASYNCcnt), then gather via LDS reads and
// write B128 stores.
// ---------------------------------------------------------------------------
__global__ __launch_bounds__(256)
void jitter_gather_kernel(const float* __restrict__ in,
                          const int4* __restrict__ idx4,
                          float* __restrict__ out) {
  __shared__ __align__(16) float tile[JITTER_T];   // 32 KB (LDS is 320 KB/WGP)

  const size_t row = blockIdx.x;
  const float* src = in + row * (size_t)JITTER_T;
  float*       dst = out + row * (size_t)JITTER_T;

  const unsigned long long sbase = (unsigned long long)(uintptr_t)src;

  // Async copy: 8 iterations x 256 threads x 16B = 32 KB.
  // VDST VGPR = wave-relative LDS byte address (low 32 bits of the generic
  // pointer to __shared__, per ISA 10.2 LDS aperture mapping).
#pragma unroll
  for (int j = 0; j < 8; ++j) {
    const unsigned elem = threadIdx.x * 4u + (unsigned)j * 1024u;  // float idx, 16B aligned
    const unsigned lds_addr = (unsigned)(uintptr_t)(tile + elem);
    const unsigned goff = elem * 4u;
    asm volatile("global_load_async_to_lds_b128 %0, %1, %2"
                 :
                 : "v"(lds_addr), "v"(goff), "s"(sbase)
                 : "memory");
  }
  asm volatile("s_wait_asynccnt 0" ::: "memory");
  __syncthreads();

  // Gather: idx table is 32 KB and shared by all 8192 blocks -> L2 resident.
#pragma unroll
  for (int j = 0; j < 8; ++j) {
    const int q = (int)threadIdx.x + j * 256;   // float4 index within row
    const int4 id = idx4[q];
    float4 v;
    v.x = tile[id.x];
    v.y = tile[id.y];
    v.z = tile[id.z];
    v.w = tile[id.w];
    reinterpret_cast<float4*>(dst)[q] = v;
  }
}

// ---------------------------------------------------------------------------
extern "C" void kernel_launch(void* const* d_in, const int* in_sizes, int n_in,
                              void* d_out, int out_size, void* d_ws, size_t ws_size,
                              hipStream_t stream) {
  (void)n_in; (void)out_size; (void)ws_size;

  const float* in = (const float*)d_in[0];
  float* out = (float*)d_out;
  int* idx = (int*)d_ws;                    // 8192 ints = 32 KB of scratch

  const int total = in_sizes[0];            // 32*256*8192
  const int rows  = total / JITTER_T;       // 8192 rows of 8192 floats

  jitter_idx_kernel<<<(HALF_T + 255) / 256, 256, 0, stream>>>(idx);
  jitter_gather_kernel<<<rows, 256, 0, stream>>>(in, (const int4*)idx, out);
}